// MessageAggregator_10677288698546
// MI455X (gfx1250) — compile-verified
//
#include <hip/hip_runtime.h>
#include <hip/hip_bf16.h>
#include <math.h>

// MI455X / gfx1250, wave32. Bandwidth-bound segment-softmax aggregation.
// WMMA (V_WMMA_F32_16X16X4_F32) used for the two small f32 GEMMs so the
// matrix math runs on the matrix pipes with exact f32 accumulation.
// B fragments are preloaded branch-free (mask-by-multiply, clamped address)
// so the inner loop is pure {A-load, WMMA} with no exec-mask churn.

typedef __attribute__((ext_vector_type(2))) float v2f;
typedef __attribute__((ext_vector_type(8))) float v8f;

#define HIDDEN 64
#define HEADS 8
#define ALPHA 0.2f
#define KSTEPS (HIDDEN / 4)  // 16 WMMAs of K=4

// ---------------------------------------------------------------------------
// Branch-free B-fragment preload for a [HEADS x HIDDEN] weight matrix used as
// B[K][Ncol] = w[Ncol][K]; columns >= HEADS are zeroed via mask multiply.
// Lane layout (f32 WMMA, wave32): Ncol = lane%16, K = 2*(lane/16) + v.
// ---------------------------------------------------------------------------
__device__ __forceinline__ void preload_b(const float* __restrict__ w,
                                          int lane, v2f bf[KSTEPS]) {
  const int Nc = lane & 15;
  const int kh = (lane >> 4) * 2;
  const int NcM = Nc & (HEADS - 1);               // clamped, always in-bounds
  const float bmask = (Nc < HEADS) ? 1.0f : 0.0f; // zero padding columns
#pragma unroll
  for (int kk = 0; kk < KSTEPS; ++kk) {
    const float* bp = w + NcM * HIDDEN + kk * 4 + kh;
    bf[kk].x = bp[0] * bmask;
    bf[kk].y = bp[1] * bmask;
  }
}

// ---------------------------------------------------------------------------
// Kernel 1: a1[n][h] = features[n,:] . attn1_w[h,:]   (N x 64) x (64 x 8)
// One wave32 handles a 16-row tile, K-loop of 16 WMMAs (K=4 each).
// ---------------------------------------------------------------------------
__global__ __launch_bounds__(256) void k_a1(const float* __restrict__ feat,
                                            const float* __restrict__ w1,
                                            float* __restrict__ a1,
                                            int nTiles, int N) {
  const int wave = (int)((blockIdx.x * blockDim.x + threadIdx.x) >> 5);
  const int lane = (int)(threadIdx.x & 31);
  if (wave >= nTiles) return;
  const int r0 = wave * 16;
  const int Mrow = lane & 15;            // A: row within tile
  const int kh = (lane >> 4) * 2;        // K sub-offset within K=4 group
  const int rowClamped = min(r0 + Mrow, N - 1);

  v2f bf[KSTEPS];
  preload_b(w1, lane, bf);

  const float* ap = feat + (long)rowClamped * HIDDEN + kh;
  v8f c = {0.f, 0.f, 0.f, 0.f, 0.f, 0.f, 0.f, 0.f};
#pragma unroll
  for (int kk = 0; kk < KSTEPS; ++kk) {
    v2f a;
    a.x = ap[kk * 4];
    a.y = ap[kk * 4 + 1];
    c = __builtin_amdgcn_wmma_f32_16x16x4_f32(false, a, false, bf[kk],
                                              (short)0, c, false, false);
  }
  // D layout: VGPR v -> M = v + 8*(lane/16), N = lane%16
  const int Nc = lane & 15;
  const int Mbase = (lane >> 4) * 8;
  if (Nc < HEADS) {
#pragma unroll
    for (int v = 0; v < 8; ++v) {
      int row = r0 + Mbase + v;
      if (row < N) a1[(long)row * HEADS + Nc] = c[v];
    }
  }
}

// ---------------------------------------------------------------------------
// Kernel 2: scores[e][h] = leakyrelu(a1[seg[e]][h] + mp[e,:] . attn2[h,:])
// Streaming 320MB pass; per-wave 16-edge tile, WMMA for the dot products.
// ---------------------------------------------------------------------------
__global__ __launch_bounds__(256) void k_scores(const float* __restrict__ mp,
                                                const float* __restrict__ w2,
                                                const float* __restrict__ a1,
                                                const int* __restrict__ seg,
                                                float* __restrict__ scores,
                                                int nTiles, int E) {
  const int wave = (int)((blockIdx.x * blockDim.x + threadIdx.x) >> 5);
  const int lane = (int)(threadIdx.x & 31);
  if (wave >= nTiles) return;
  const long e0 = (long)wave * 16;
  const int Mrow = lane & 15;
  const int kh = (lane >> 4) * 2;
  const long rowClamped = min(e0 + Mrow, (long)E - 1);

  v2f bf[KSTEPS];
  preload_b(w2, lane, bf);

  const float* ap = mp + rowClamped * HIDDEN + kh;
  v8f c = {0.f, 0.f, 0.f, 0.f, 0.f, 0.f, 0.f, 0.f};
#pragma unroll
  for (int kk = 0; kk < KSTEPS; ++kk) {
    v2f a;
    a.x = ap[kk * 4];
    a.y = ap[kk * 4 + 1];
    c = __builtin_amdgcn_wmma_f32_16x16x4_f32(false, a, false, bf[kk],
                                              (short)0, c, false, false);
  }
  const int Nc = lane & 15;
  const int Mbase = (lane >> 4) * 8;
  if (Nc < HEADS) {
#pragma unroll
    for (int v = 0; v < 8; ++v) {
      long e = e0 + Mbase + v;
      if (e < (long)E) {
        int node = seg[e];
        float s = a1[(long)node * HEADS + Nc] + c[v];
        s = (s > 0.f) ? s : ALPHA * s;  // leaky relu
        scores[e * HEADS + Nc] = s;
      }
    }
  }
}

// ---------------------------------------------------------------------------
// Kernel 3: CSR offsets via binary search (segment_ids is sorted).
// row_start[n] = first edge with seg >= n ; row_start[N] = E.
// ---------------------------------------------------------------------------
__global__ void k_rowstart(const int* __restrict__ seg,
                           int* __restrict__ row_start, int N, int E) {
  int n = (int)(blockIdx.x * blockDim.x + threadIdx.x);
  if (n > N) return;
  int lo = 0, hi = E;
  while (lo < hi) {
    int mid = (lo + hi) >> 1;
    if (seg[mid] < n) lo = mid + 1; else hi = mid;
  }
  row_start[n] = lo;
}

// ---------------------------------------------------------------------------
// Kernel 4: per-node stable softmax + weighted aggregation + ELU.
// One 256-thread block per node; no atomics; coalesced output writes.
// Thread t owns output dims (h0 = t/64, d = t%64) and (h0+4, d).
// ---------------------------------------------------------------------------
__global__ __launch_bounds__(256) void k_aggregate(
    const float* __restrict__ mp, const float* __restrict__ scores,
    const int* __restrict__ row_start, float* __restrict__ out) {
  __shared__ float red[256];
  __shared__ float attn[64 * HEADS];
  __shared__ float mh[HEADS];
  __shared__ float dh[HEADS];

  const int n = (int)blockIdx.x;
  const int t = (int)threadIdx.x;
  const int s = row_start[n];
  const int e = row_start[n + 1];
  const int h0 = t >> 6;    // 0..3
  const int d = t & 63;
  float* o = out + (long)n * (HEADS * HIDDEN);

  if (s == e) {  // empty segment -> elu(0) = 0
    o[h0 * HIDDEN + d] = 0.f;
    o[(h0 + 4) * HIDDEN + d] = 0.f;
    return;
  }

  const int h = t & 7;
  const int g = t >> 3;  // 32 strided groups per head

  // ---- per-head max ----
  float lm = -__builtin_inff();
  for (int i = s + g; i < e; i += 32) lm = fmaxf(lm, scores[(long)i * HEADS + h]);
  red[t] = lm;
  __syncthreads();
  for (int st = 128; st >= 8; st >>= 1) {
    if (t < st) red[t] = fmaxf(red[t], red[t + st]);
    __syncthreads();
  }
  if (t < HEADS) mh[t] = red[t];
  __syncthreads();

  // ---- per-head exp-sum ----
  const float mhv = mh[h];
  float ls = 0.f;
  for (int i = s + g; i < e; i += 32) ls += expf(scores[(long)i * HEADS + h] - mhv);
  __syncthreads();
  red[t] = ls;
  __syncthreads();
  for (int st = 128; st >= 8; st >>= 1) {
    if (t < st) red[t] += red[t + st];
    __syncthreads();
  }
  if (t < HEADS) dh[t] = red[t];  // >= 1 since max element contributes exp(0)
  __syncthreads();

  // ---- chunked weighted aggregation ----
  float acc0 = 0.f, acc1 = 0.f;
  for (int c0 = s; c0 < e; c0 += 64) {
    const int cnt = min(64, e - c0);
    __syncthreads();
    for (int idx = t; idx < cnt * HEADS; idx += 256) {
      const int ii = idx >> 3, hh = idx & 7;
      attn[idx] = expf(scores[(long)(c0 + ii) * HEADS + hh] - mh[hh]) / dh[hh];
    }
    __syncthreads();
    for (int ii = 0; ii < cnt; ++ii) {
      const float mv = mp[(long)(c0 + ii) * HIDDEN + d];
      acc0 = fmaf(attn[ii * HEADS + h0], mv, acc0);
      acc1 = fmaf(attn[ii * HEADS + h0 + 4], mv, acc1);
    }
  }

  o[h0 * HIDDEN + d] = (acc0 > 0.f) ? acc0 : expm1f(acc0);
  o[(h0 + 4) * HIDDEN + d] = (acc1 > 0.f) ? acc1 : expm1f(acc1);
}

// ---------------------------------------------------------------------------
// Host launcher
// ---------------------------------------------------------------------------
extern "C" void kernel_launch(void* const* d_in, const int* in_sizes, int n_in,
                              void* d_out, int out_size, void* d_ws, size_t ws_size,
                              hipStream_t stream) {
  const float* feat = (const float*)d_in[0];   // [N, 64]
  const float* mp   = (const float*)d_in[1];   // [E, 64]
  const float* w1   = (const float*)d_in[2];   // [8, 64]
  const float* w2   = (const float*)d_in[3];   // [8, 64]
  const int*   seg  = (const int*)d_in[4];     // [E], sorted

  const int N = in_sizes[0] / HIDDEN;
  const int E = in_sizes[1] / HIDDEN;
  float* out = (float*)d_out;

  // Workspace layout: a1 [N*8] f32 | scores [E*8] f32 | row_start [N+1] i32
  char* ws = (char*)d_ws;
  float* a1 = (float*)ws;
  float* scores = (float*)(ws + (size_t)N * HEADS * sizeof(float));
  int* row_start = (int*)(ws + (size_t)N * HEADS * sizeof(float)
                             + (size_t)E * HEADS * sizeof(float));

  const int t1 = (N + 15) / 16;  // 16-row tiles
  k_a1<<<(t1 + 7) / 8, 256, 0, stream>>>(feat, w1, a1, t1, N);

  const int t2 = (E + 15) / 16;  // 16-edge tiles
  k_scores<<<(t2 + 7) / 8, 256, 0, stream>>>(mp, w2, a1, seg, scores, t2, E);

  k_rowstart<<<(N + 1 + 255) / 256, 256, 0, stream>>>(seg, row_start, N, E);

  k_aggregate<<<N, 256, 0, stream>>>(mp, scores, row_start, out);
}